// DenseToSparse_49022756717275
// MI455X (gfx1250) — compile-verified
//
#include <hip/hip_runtime.h>
#include <stdint.h>

// Problem constants (B,C,H,W) = (32,64,128,128)
#define Cn      64
#define HWn     16384            // H*W
#define Sn      524288           // B*H*W sites
#define TILE    128              // sites per tile
#define NTILES  (Sn / TILE)      // 4096
#define LSTRIDE 129              // LDS row stride in floats == TDM tile_dim0 (odd -> no bank conflicts)

typedef unsigned int u32;
typedef unsigned long long u64;
typedef __attribute__((ext_vector_type(4))) u32 u32x4_t;
typedef __attribute__((ext_vector_type(8))) int i32x8_t;
typedef __attribute__((ext_vector_type(4))) int i32x4_t;

#if __has_builtin(__builtin_amdgcn_tensor_load_to_lds)
#define HAVE_TDM 1
#else
#define HAVE_TDM 0
#endif

__device__ __forceinline__ void wait_tensorcnt0() {
#if __has_builtin(__builtin_amdgcn_s_wait_tensorcnt)
  __builtin_amdgcn_s_wait_tensorcnt(0);
#else
  asm volatile("s_wait_tensorcnt 0" ::: "memory");
#endif
}

// TDM: async-load `nrows` rows of LSTRIDE floats (memory row stride = HWn floats)
// packed back-to-back into LDS. tensor_dim0 = td0 bounds the fast axis so the
// 129th element of the last tile in a row is OOB-zeroed, never read past x.
__device__ __forceinline__ void tdm_load_rows(const float* gp, u32 ldsOff, u32 td0, u32 nrows) {
#if HAVE_TDM
  const u64 ga = (u64)(uintptr_t)gp;
  u32x4_t g0;
  g0.x = 1u;                                               // count=1 (valid), user mode
  g0.y = ldsOff;                                           // LDS byte address
  g0.z = (u32)ga;                                          // global_addr[31:0]
  g0.w = ((u32)((ga >> 32) & 0x01FFFFFFull)) | (2u << 30); // addr[56:32] | type=2
  i32x8_t g1;
  g1[0] = (int)(2u << 16);                                 // wg_mask=0, data_size=4B, no pad/iter
  g1[1] = (int)((td0 & 0xFFFFu) << 16);                    // abar=0, tensor_dim0[15:0]
  g1[2] = (int)((td0 >> 16) | ((nrows & 0xFFFFu) << 16));  // tensor_dim0[31:16], tensor_dim1
  g1[3] = (int)((u32)LSTRIDE << 16);                       // tile_dim0 = 129
  g1[4] = (int)(nrows & 0xFFFFu);                          // tile_dim1 = nrows, tile_dim2 = 0
  g1[5] = (int)(u32)HWn;                                   // tensor_dim0_stride = 16384 elems
  g1[6] = 0;
  g1[7] = 0;
  const i32x4_t z4 = {0, 0, 0, 0};
#if defined(__clang_major__) && (__clang_major__ >= 23)
  const i32x8_t z8 = {0, 0, 0, 0, 0, 0, 0, 0};
  __builtin_amdgcn_tensor_load_to_lds(g0, g1, z4, z4, z8, 0);
#else
  __builtin_amdgcn_tensor_load_to_lds(g0, g1, z4, z4, 0);
#endif
#else
  (void)gp; (void)ldsOff; (void)td0; (void)nrows;
#endif
}

// ---------------- K1: per-tile active counts. One wave per 128-site tile,
// float4 per lane -> one b128 load per channel per tile. No LDS, no barriers.
__global__ __launch_bounds__(256) void k_flags(const float* __restrict__ x,
                                               u32* __restrict__ sums) {
  const int t = threadIdx.x;
  const int wv = t >> 5, lane = t & 31;
  const int tileId = blockIdx.x * 8 + wv;
  const int s0 = tileId * TILE;
  const int b = s0 >> 14;
  const int hw0 = s0 & (HWn - 1);
  const float4* p = (const float4*)(x + ((size_t)b << 20) + (size_t)hw0) + lane;
  bool a0 = false, a1 = false, a2 = false, a3 = false;
#pragma unroll 16
  for (int c = 0; c < Cn; ++c) {
    const float4 v = p[(size_t)c * (HWn / 4)];
    a0 |= (v.x != 0.0f);
    a1 |= (v.y != 0.0f);
    a2 |= (v.z != 0.0f);
    a3 |= (v.w != 0.0f);
  }
  const u32 cnt = (u32)__popc(__builtin_amdgcn_ballot_w32(a0)) +
                  (u32)__popc(__builtin_amdgcn_ballot_w32(a1)) +
                  (u32)__popc(__builtin_amdgcn_ballot_w32(a2)) +
                  (u32)__popc(__builtin_amdgcn_ballot_w32(a3));
  if (lane == 0) sums[tileId] = cnt;
}

// ---------------- K2: exclusive scan of 4096 tile counts (one block)
__global__ __launch_bounds__(256) void k_scan(const u32* __restrict__ sums,
                                              u32* __restrict__ offs,
                                              u32* __restrict__ cnt,
                                              float* dcount) {
  __shared__ u32 sh[256];
  const int t = threadIdx.x;
  u32 loc[16];
  u32 tot = 0;
  const int base = t * 16;
#pragma unroll
  for (int i = 0; i < 16; ++i) { loc[i] = sums[base + i]; tot += loc[i]; }
  sh[t] = tot;
  __syncthreads();
  u32 incl = tot;
  for (int off = 1; off < 256; off <<= 1) {
    const u32 v = (t >= off) ? sh[t - off] : 0u;
    __syncthreads();
    incl += v;
    sh[t] = incl;
    __syncthreads();
  }
  u32 run = incl - tot;                                    // exclusive base for this thread
#pragma unroll
  for (int i = 0; i < 16; ++i) { offs[base + i] = run; run += loc[i]; }
  if (t == 255) {
    cnt[0] = incl;                                         // total active sites
    if (dcount) dcount[0] = (float)incl;
  }
}

// ---------------- K3: TDM gather -> LDS transpose -> compacted row scatter
__global__ __launch_bounds__(256) void k_scatter(const float* __restrict__ x,
                                                 float* __restrict__ out,
                                                 const u32* __restrict__ offs,
                                                 const u32* __restrict__ cntPtr) {
  __shared__ float tile[Cn * LSTRIDE];                     // 64 x 129 floats (33 KB)
  __shared__ u32 destRow[TILE];
  __shared__ u32 pflag[256];
  __shared__ u32 wsum[8];

  const int tid = threadIdx.x;
  const int tileId = blockIdx.x;
  const int s0 = tileId * TILE;
  const int b = s0 >> 14;
  const int hw0 = s0 & (HWn - 1);
  const int lane = tid & 31, wv = tid >> 5;

#if HAVE_TDM
  {
    // each of the 8 waves DMAs its 8 channel-rows into LDS (async, TENSORcnt)
    const float* gp = x + ((size_t)b << 20) + ((size_t)(wv * 8) << 14) + (size_t)hw0;
    const u32 ldsOff = (u32)(uintptr_t)tile + (u32)(wv * 8 * LSTRIDE * 4);
    tdm_load_rows(gp, ldsOff, (u32)(HWn - hw0), 8u);
    wait_tensorcnt0();
  }
#else
  for (int i = tid; i < Cn * TILE; i += 256) {
    const int c = i >> 7;
    const int ss = i & (TILE - 1);
    tile[c * LSTRIDE + ss] = x[((size_t)b << 20) + ((size_t)c << 14) + (size_t)(hw0 + ss)];
  }
#endif
  __syncthreads();

  // per-site active flag: all 8 waves participate (each thread ORs 32 channels)
  {
    const int site = tid & (TILE - 1);
    const int cbase = (tid >> 7) * 32;                     // 0 or 32
    bool ap = false;
#pragma unroll 8
    for (int c = 0; c < 32; ++c)
      ap |= (tile[(cbase + c) * LSTRIDE + site] != 0.0f);
    pflag[tid] = ap ? 1u : 0u;
  }
  __syncthreads();

  bool act = false;
  if (tid < TILE) act = (pflag[tid] | pflag[tid + TILE]) != 0u;
  const unsigned mask = __builtin_amdgcn_ballot_w32(act);  // convergent, all 8 waves
  if (lane == 0) wsum[wv] = (u32)__popc(mask);
  __syncthreads();

  if (tid < TILE) {
    u32 wbase = 0;
    for (int w = 0; w < wv; ++w) wbase += wsum[w];
    const u32 localRank = wbase + (u32)__popc(mask & ((1u << lane) - 1u));
    const u32 activeBefore = offs[tileId] + localRank;     // actives strictly before this site
    const u32 sGlobal = (u32)(s0 + tid);
    const u32 total = cntPtr[0];
    // active row -> its compacted slot; inactive row -> zero row in the tail
    // (inactive rows are all-zero in LDS already, so no masking is needed)
    destRow[tid] = act ? activeBefore : (total + (sGlobal - activeBefore));
  }
  __syncthreads();

  // transpose write: 16 lanes x float4 per row -> 256B contiguous stores.
  // LDS reads stay conflict-free because LSTRIDE=129 is odd.
  const int ch4 = (tid & 15) << 2;                         // 0,4,...,60
  const int sgrp = tid >> 4;                               // 16 rows in flight
#pragma unroll
  for (int it = 0; it < TILE / 16; ++it) {                 // 8 iterations
    const int site = (it << 4) + sgrp;
    float4 v;
    v.x = tile[(ch4 + 0) * LSTRIDE + site];
    v.y = tile[(ch4 + 1) * LSTRIDE + site];
    v.z = tile[(ch4 + 2) * LSTRIDE + site];
    v.w = tile[(ch4 + 3) * LSTRIDE + site];
    *(float4*)(out + (size_t)destRow[site] * Cn + ch4) = v;
  }
}

extern "C" void kernel_launch(void* const* d_in, const int* in_sizes, int n_in,
                              void* d_out, int out_size, void* d_ws, size_t ws_size,
                              hipStream_t stream) {
  (void)in_sizes; (void)n_in; (void)ws_size;
  const float* x = (const float*)d_in[0];
  float* out = (float*)d_out;

  u32* sums = (u32*)d_ws;           // [NTILES]
  u32* offs = sums + NTILES;        // [NTILES]
  u32* cnt  = offs + NTILES;        // [1]

  float* dcount = (out_size > (int)((size_t)Sn * Cn)) ? (out + (size_t)Sn * Cn) : nullptr;

  k_flags  <<<NTILES / 8, 256, 0, stream>>>(x, sums);
  k_scan   <<<1,          256, 0, stream>>>(sums, offs, cnt, dcount);
  k_scatter<<<NTILES,     256, 0, stream>>>(x, out, offs, cnt);
}